// Capsule_455266534023
// MI455X (gfx1250) — compile-verified
//
#include <hip/hip_runtime.h>
#include <hip/hip_bf16.h>

#define B_   16
#define NIN  512
#define DIN  256      // K of the GEMM
#define NC   64       // NUM_CAPSULE
#define DC   128      // DIM_CAPSULE
#define ROUTINGS 3

typedef __attribute__((ext_vector_type(2))) float v2f;
typedef __attribute__((ext_vector_type(8))) float v8f;

// ---------------------------------------------------------------------------
// Kernel 1: hat[b][c][n][d] = x[b] @ kernel[:, c*DC : (c+1)*DC]
// One workgroup per (b,c). 256 threads = 8 waves.
// W_c (256x128 f32 = 128KB) staged in LDS with k-pair swizzle:
//   wlds[(k>>1)*256 + n*2 + (k&1)]  -> B fragment = one aligned 8-byte LDS load
// Each wave: 4 M-tiles of 16 rows, 8 fp32 WMMA accumulators covering D=128.
// ---------------------------------------------------------------------------
__global__ void __launch_bounds__(256)
gemm_hat_wmma(const float* __restrict__ x, const float* __restrict__ w,
              float* __restrict__ hat)
{
    extern __shared__ float wlds[];                  // 32768 floats = 128 KB
    const int c = (int)blockIdx.x;
    const int b = (int)blockIdx.y;
    const int t = (int)threadIdx.x;

    // Cooperative, coalesced load of W_c into LDS (float4 along n).
    const float* wg = w + (size_t)c * DC;            // row stride NC*DC = 8192
    #pragma unroll
    for (int i = 0; i < 32; ++i) {
        const int flat = i * 256 + t;                // 8192 float4 total
        const int k = flat >> 5;                     // 32 float4 per k-row
        const int n = (flat & 31) << 2;
        const float4 v = *reinterpret_cast<const float4*>(wg + (size_t)k * (NC * DC) + n);
        float* dst = &wlds[(k >> 1) * (2 * DC) + (k & 1)];
        dst[(n + 0) * 2] = v.x;
        dst[(n + 1) * 2] = v.y;
        dst[(n + 2) * 2] = v.z;
        dst[(n + 3) * 2] = v.w;
    }
    __syncthreads();

    const int lane   = t & 31;
    const int wv     = t >> 5;
    const int laneLo = lane & 15;
    const int laneHi = lane >> 4;

    const float* xb = x + (size_t)b * NIN * DIN;

    for (int mt = 0; mt < 4; ++mt) {
        const int m0 = (wv * 4 + mt) * 16;           // 8 waves * 4 tiles * 16 = 512 rows
        // A fragment source: x[b][m0+laneLo][k + 2*laneHi + {0,1}]
        const float* xa = xb + (size_t)(m0 + laneLo) * DIN + 2 * laneHi;

        v8f acc[8] = {};
        for (int k = 0; k < DIN; k += 4) {
            const v2f a = *reinterpret_cast<const v2f*>(xa + k);
            const float* bbase = &wlds[((k >> 1) + laneHi) * (2 * DC) + laneLo * 2];
            #pragma unroll
            for (int nt = 0; nt < 8; ++nt) {
                const v2f bf = *reinterpret_cast<const v2f*>(bbase + nt * 32);
                acc[nt] = __builtin_amdgcn_wmma_f32_16x16x4_f32(
                    false, a, false, bf, (short)0, acc[nt], false, false);
            }
        }

        // C/D layout: VGPR r holds row (m0 + r + 8*laneHi), col = laneLo (+16*nt)
        float* hb = hat + (((size_t)(b * NC + c) * NIN) + m0 + laneHi * 8) * DC + laneLo;
        #pragma unroll
        for (int nt = 0; nt < 8; ++nt) {
            #pragma unroll
            for (int r = 0; r < 8; ++r)
                hb[(size_t)r * DC + nt * 16] = acc[nt][r];
        }
    }
}

// ---------------------------------------------------------------------------
// Kernel 2: per-(b,n) softmax statistics across the 64 capsules.
// blog layout: [b][c][n]. Tiny (2 MB), fully L2 resident.
// ---------------------------------------------------------------------------
__global__ void __launch_bounds__(256)
softmax_prep(const float* __restrict__ blog, float* __restrict__ mx,
             float* __restrict__ denom)
{
    const int tid = (int)(blockIdx.x * blockDim.x + threadIdx.x);
    if (tid >= B_ * NIN) return;
    const int b = tid / NIN;
    const int n = tid - b * NIN;
    const float* p = blog + (size_t)b * NC * NIN + n;
    float m = -3.402823466e38f;
    for (int c = 0; c < NC; ++c) m = fmaxf(m, p[(size_t)c * NIN]);
    float s = 0.f;
    for (int c = 0; c < NC; ++c) s += __expf(p[(size_t)c * NIN] - m);
    mx[tid] = m;
    denom[tid] = s;
}

// ---------------------------------------------------------------------------
// Kernel 3: one routing iteration. One workgroup per (b,c).
// hat[b,c,:,:] (256 KB) cached in LDS once; both the o-reduction and the
// logit update read from LDS, halving global hat traffic.
// iter==0: c = 1/64 uniform (softmax of zeros), blog written fresh.
// ---------------------------------------------------------------------------
__global__ void __launch_bounds__(256)
routing_iter(const float* __restrict__ hat, float* __restrict__ blog,
             const float* __restrict__ mx, const float* __restrict__ denom,
             float* __restrict__ out, int iter)
{
    extern __shared__ float hat_s[];                 // NIN*DC floats = 256 KB
    __shared__ float c_s[NIN];
    __shared__ float part_s[256];
    __shared__ float o_s[DC];
    __shared__ float scale_s;

    const int c = (int)blockIdx.x;
    const int b = (int)blockIdx.y;
    const int t = (int)threadIdx.x;
    const size_t bc = (size_t)b * NC + c;

    // Stage hat slice into LDS (coalesced float4).
    const float4* hg4 = reinterpret_cast<const float4*>(hat + bc * (NIN * DC));
    float4* hs4 = reinterpret_cast<float4*>(hat_s);
    #pragma unroll
    for (int i = 0; i < (NIN * DC / 4) / 256; ++i)
        hs4[i * 256 + t] = hg4[i * 256 + t];

    // Coupling coefficients c[n] = softmax_c(blog)[this c]
    for (int n = t; n < NIN; n += 256) {
        float cc;
        if (iter == 0) {
            cc = 1.0f / (float)NC;
        } else {
            const float lv = blog[bc * NIN + n];
            cc = __expf(lv - mx[b * NIN + n]) / denom[b * NIN + n];
        }
        c_s[n] = cc;
    }
    __syncthreads();

    // s[d] = sum_n c[n] * hat[n][d]  (two half-threads per d)
    const int d    = t & (DC - 1);
    const int half = t >> 7;
    float acc = 0.f;
    for (int n = half; n < NIN; n += 2)
        acc += c_s[n] * hat_s[n * DC + d];
    part_s[t] = acc;
    __syncthreads();
    if (t < DC) o_s[t] = part_s[t] + part_s[t + DC];
    __syncthreads();

    // squash: scale = sqrt(qn) / (0.5 + qn), qn = |s|^2 + 1e-7
    if (t == 0) {
        float qn = 1e-7f;
        for (int i = 0; i < DC; ++i) qn += o_s[i] * o_s[i];
        scale_s = __fsqrt_rn(qn) / (0.5f + qn);
    }
    __syncthreads();
    if (t < DC) o_s[t] *= scale_s;
    __syncthreads();

    if (iter == ROUTINGS - 1) {
        if (t < DC) out[bc * DC + t] = o_s[t];
    } else {
        // blog[n] += dot(o, hat[n,:]); bank-stagger the d walk.
        #pragma unroll
        for (int j = 0; j < 2; ++j) {
            const int n = t * 2 + j;
            const int off = (t & 31) * 4;
            float dot = 0.f;
            for (int dd = 0; dd < DC; ++dd) {
                const int di = (dd + off) & (DC - 1);
                dot += o_s[di] * hat_s[n * DC + di];
            }
            const float prev = (iter == 0) ? 0.f : blog[bc * NIN + n];
            blog[bc * NIN + n] = prev + dot;
        }
    }
}

// ---------------------------------------------------------------------------
extern "C" void kernel_launch(void* const* d_in, const int* in_sizes, int n_in,
                              void* d_out, int out_size, void* d_ws, size_t ws_size,
                              hipStream_t stream)
{
    (void)in_sizes; (void)n_in; (void)out_size; (void)ws_size;
    const float* x = (const float*)d_in[0];
    const float* w = (const float*)d_in[1];
    float* out = (float*)d_out;

    // Workspace layout
    float* hat  = (float*)d_ws;                                   // 256 MB
    float* blog = hat + (size_t)B_ * NC * NIN * DC;               // 2 MB
    float* mxv  = blog + (size_t)B_ * NC * NIN;                   // 32 KB
    float* dnv  = mxv + (size_t)B_ * NIN;                         // 32 KB

    const dim3 grid(NC, B_);
    const size_t wshm = (size_t)(DIN / 2) * (2 * DC) * sizeof(float); // 128 KB
    const size_t hshm = (size_t)NIN * DC * sizeof(float);             // 256 KB

    gemm_hat_wmma<<<grid, 256, wshm, stream>>>(x, w, hat);

    routing_iter<<<grid, 256, hshm, stream>>>(hat, blog, mxv, dnv, out, 0);
    softmax_prep<<<(B_ * NIN + 255) / 256, 256, 0, stream>>>(blog, mxv, dnv);
    routing_iter<<<grid, 256, hshm, stream>>>(hat, blog, mxv, dnv, out, 1);
    softmax_prep<<<(B_ * NIN + 255) / 256, 256, 0, stream>>>(blog, mxv, dnv);
    routing_iter<<<grid, 256, hshm, stream>>>(hat, blog, mxv, dnv, out, 2);
}